// chamfer_3DDist_1116691497030
// MI455X (gfx1250) — compile-verified
//
#include <hip/hip_runtime.h>

typedef float v2f __attribute__((ext_vector_type(2)));
typedef float v8f __attribute__((ext_vector_type(8)));

#define BATCH 16
#define NPTS  4096
#define MPTS  4096

// Nearest-neighbor of `query` points against `ref` points using
// V_WMMA_F32_16X16X4_F32 for the -2*x.y + |y|^2 tile computation.
//
// Grid: (Nq/128, B). Block: 256 threads = 8 waves; each wave owns 16 queries.
// LDS: all Nr refs staged as float4 (y0,y1,y2,|y|^2) = 64KB.
__global__ __launch_bounds__(256)
void nn_wmma_kernel(const float* __restrict__ query,
                    const float* __restrict__ ref,
                    float* __restrict__ dist_out,
                    int*   __restrict__ idx_out,
                    int Nq, int Nr)
{
    __shared__ float4 s_ref[MPTS];

    const int b    = blockIdx.y;
    const int tid  = threadIdx.x;
    const int lane = tid & 31;
    const int wave = tid >> 5;

    // Stage refs into LDS with precomputed squared norm.
    const float* refB = ref + (size_t)b * Nr * 3;
    for (int i = tid; i < Nr; i += 256) {
        float y0 = refB[3 * i + 0];
        float y1 = refB[3 * i + 1];
        float y2 = refB[3 * i + 2];
        s_ref[i] = make_float4(y0, y1, y2, y0 * y0 + y1 * y1 + y2 * y2);
    }
    __syncthreads();

    const int qbase = blockIdx.x * 128 + wave * 16;
    const float* qB = query + (size_t)b * Nq * 3;

    // Each lane loads query point q = qbase + (lane & 15).
    const int ql = qbase + (lane & 15);
    float x0 = qB[3 * ql + 0];
    float x1 = qB[3 * ql + 1];
    float x2 = qB[3 * ql + 2];
    float xnorm = x0 * x0 + x1 * x1 + x2 * x2;

    // A fragment (16x4 f32): row q = [-2x0, -2x1, -2x2, 1]
    // lanes 0-15: K=0,1 ; lanes 16-31: K=2,3   (select once, outside the loop)
    v2f a;
    if (lane < 16) { a.x = -2.0f * x0; a.y = -2.0f * x1; }
    else           { a.x = -2.0f * x2; a.y = 1.0f;       }

    float best[8];
    int   bidx[8];
#pragma unroll
    for (int r = 0; r < 8; ++r) { best[r] = __builtin_inff(); bidx[r] = 0; }

    // Branchless B-fragment fetch: view each float4 ref entry as two float2
    // halves; lanes 0-15 read (y0,y1) [half 0], lanes 16-31 read (y2,|y|^2)
    // [half 1]. One ds_load_b64 per tile, no divergent address computation.
    const float2* s2  = (const float2*)s_ref;
    const int ml      = lane & 15;
    const int halfsel = (lane >> 4) & 1;          // 0: lanes 0-15, 1: lanes 16-31
    const int ebase   = 2 * ml + halfsel;         // float2-element index at mt=0

    // Two 16-ref tiles per iteration: second tile's load overlaps the first
    // tile's argmin VALU work.
    for (int mt = 0; mt < Nr; mt += 32) {
        float2 ta = s2[2 * mt + ebase];
        float2 tb = s2[2 * (mt + 16) + ebase];

        v2f bfa; bfa.x = ta.x; bfa.y = ta.y;
        v2f bfb; bfb.x = tb.x; bfb.y = tb.y;

        v8f c0 = {};
        v8f c1 = {};
        // D[q, m] = -2 * x_q . y_m + |y_m|^2  (= d(q,m) - |x_q|^2)
        c0 = __builtin_amdgcn_wmma_f32_16x16x4_f32(
                false, a, false, bfa, (short)0, c0, false, false);
        c1 = __builtin_amdgcn_wmma_f32_16x16x4_f32(
                false, a, false, bfb, (short)0, c1, false, false);

        const int mi0 = mt + ml;
        const int mi1 = mt + 16 + ml;
#pragma unroll
        for (int r = 0; r < 8; ++r) {
            float d0 = c0[r];
            if (d0 < best[r]) { best[r] = d0; bidx[r] = mi0; }  // strict < keeps first index
        }
#pragma unroll
        for (int r = 0; r < 8; ++r) {
            float d1 = c1[r];
            if (d1 < best[r]) { best[r] = d1; bidx[r] = mi1; }
        }
    }

    // Reduce across the 16 lanes of each half (masks 1,2,4,8 stay within halves).
#pragma unroll
    for (int m = 1; m <= 8; m <<= 1) {
#pragma unroll
        for (int r = 0; r < 8; ++r) {
            float ob = __shfl_xor(best[r], m, 32);
            int   oi = __shfl_xor(bidx[r], m, 32);
            if (ob < best[r] || (ob == best[r] && oi < bidx[r])) {
                best[r] = ob;
                bidx[r] = oi;
            }
        }
    }

    // C/D layout: lanes 0-15 VGPR r = row r; lanes 16-31 VGPR r = row 8+r.
    // Lane r (r<8) holds query row r; lane 24+r holds query row 8+r.
    const size_t obase = (size_t)b * Nq + qbase;
#pragma unroll
    for (int r = 0; r < 8; ++r) {
        if (lane == r) {
            dist_out[obase + r] = best[r] + xnorm;
            idx_out [obase + r] = bidx[r];
        }
        if (lane == 24 + r) {
            dist_out[obase + 8 + r] = best[r] + xnorm;
            idx_out [obase + 8 + r] = bidx[r];
        }
    }
}

extern "C" void kernel_launch(void* const* d_in, const int* in_sizes, int n_in,
                              void* d_out, int out_size, void* d_ws, size_t ws_size,
                              hipStream_t stream) {
    const float* xyz1 = (const float*)d_in[0];  // [B, N, 3]
    const float* xyz2 = (const float*)d_in[1];  // [B, M, 3]

    float* out   = (float*)d_out;
    float* dist1 = out;                                       // [B, N]
    float* dist2 = out + (size_t)BATCH * NPTS;                // [B, M]
    int*   idx1  = (int*)(out + (size_t)BATCH * NPTS + (size_t)BATCH * MPTS);
    int*   idx2  = idx1 + (size_t)BATCH * NPTS;

    dim3 block(256);
    dim3 grid1(NPTS / 128, BATCH);
    nn_wmma_kernel<<<grid1, block, 0, stream>>>(xyz1, xyz2, dist1, idx1, NPTS, MPTS);
    dim3 grid2(MPTS / 128, BATCH);
    nn_wmma_kernel<<<grid2, block, 0, stream>>>(xyz2, xyz1, dist2, idx2, MPTS, NPTS);
}